// FCOS_eval_89421219103630
// MI455X (gfx1250) — compile-verified
//
#include <hip/hip_runtime.h>
#include <stdint.h>

#define NB    16
#define NCLS  80
#define NTOT  17064
#define KTOP  1000
#define SORTN 32768
#define SCORE_THR 0.05f
#define IOU_THR   0.6f

#define AS1 __attribute__((address_space(1)))
#define AS3 __attribute__((address_space(3)))

// ---------------- CDNA5 async global->LDS staging -------------------------
#if defined(__has_builtin)
#if __has_builtin(__builtin_amdgcn_global_load_async_to_lds_b32)
#define HAVE_ASYNC_BUILTIN 1
#endif
#if __has_builtin(__builtin_amdgcn_s_wait_asynccnt)
#define HAVE_WAIT_BUILTIN 1
#endif
#endif

// Stage 4 bytes per lane from global to LDS (async, tracked by ASYNCcnt).
// gbase must be wave-uniform; voff_bytes is the per-lane byte offset.
__device__ __forceinline__ void stage_b32(float* lds_dst, const float* gbase,
                                          unsigned voff_bytes) {
#if defined(HAVE_ASYNC_BUILTIN)
  // inttoptr casts sidestep addrspace-cast legality questions:
  // low 32 bits of a flat LDS pointer are the LDS byte offset.
  __builtin_amdgcn_global_load_async_to_lds_b32(
      (AS1 int*)(size_t)((const char*)gbase + voff_bytes),
      (AS3 int*)(size_t)(unsigned)(size_t)lds_dst,
      0, 0);
#else
  unsigned lds = (unsigned)(size_t)lds_dst;  // LDS byte offset (low 32 bits of flat)
  asm volatile("global_load_async_to_lds_b32 %0, %1, %2 offset:0"
               :: "v"(lds), "v"(voff_bytes), "s"(gbase)
               : "memory");
#endif
}

__device__ __forceinline__ void wait_async_le1() {
#if defined(HAVE_WAIT_BUILTIN)
  __builtin_amdgcn_s_wait_asynccnt(1);
#else
  asm volatile("s_wait_asynccnt 0x1" ::: "memory");
#endif
}
__device__ __forceinline__ void wait_async_le0() {
#if defined(HAVE_WAIT_BUILTIN)
  __builtin_amdgcn_s_wait_asynccnt(0);
#else
  asm volatile("s_wait_asynccnt 0x0" ::: "memory");
#endif
}

// ---------------- Phase 1: per-anchor decode ------------------------------
// score = exp((max_c cls + cen)/2), class = argmax_c cls, box = coord -/+ exp(reg)
__global__ void decode_kernel(const float* __restrict__ cls,
                              const float* __restrict__ cen,
                              const float* __restrict__ reg,
                              float* __restrict__ scores,
                              int*   __restrict__ klass,
                              float* __restrict__ boxes,
                              int HW, int W, int stride, int aoff, int tilesPerImg) {
  __shared__ float sbuf[2][256];
  const int b   = blockIdx.x / tilesPerImg;
  const int hw0 = (blockIdx.x % tilesPerImg) * 256;
  const int tid = threadIdx.x;
  const int hw  = hw0 + tid;
  const int hwc = hw < HW ? hw : HW - 1;        // clamp so every lane issues in-bounds
  const float* cbase = cls + (size_t)b * NCLS * HW;
  const unsigned voff = (unsigned)hwc * 4u;

  stage_b32(&sbuf[0][tid], cbase, voff);        // prime channel 0
  float best = -3.4e38f; int besti = 0;
  for (int c = 0; c < NCLS; ++c) {
    if (c + 1 < NCLS) {
      stage_b32(&sbuf[(c + 1) & 1][tid], cbase + (size_t)(c + 1) * HW, voff);
      wait_async_le1();                         // channel c complete (in-order)
    } else {
      wait_async_le0();
    }
    float v = sbuf[c & 1][tid];
    if (v > best) { best = v; besti = c; }
  }
  if (hw < HW) {
    float ce = cen[(size_t)b * HW + hw];
    float sc = __expf(0.5f * (best + ce));
    const float* rb = reg + (size_t)b * 4 * HW + hw;
    float r0 = __expf(rb[0]);
    float r1 = __expf(rb[(size_t)HW]);
    float r2 = __expf(rb[(size_t)2 * HW]);
    float r3 = __expf(rb[(size_t)3 * HW]);
    int h = hw / W, w = hw - h * W;
    float cx = (float)(w * stride + (stride >> 1));
    float cy = (float)(h * stride + (stride >> 1));
    int g = b * NTOT + aoff + hw;
    scores[g] = sc;
    klass[g]  = besti;
    float* bx = boxes + (size_t)g * 4;
    bx[0] = cx - r0; bx[1] = cy - r1; bx[2] = cx + r2; bx[3] = cy + r3;
  }
}

// ---------------- Phase 2: per-image sorted top-K (bitonic, L2-resident) --
__global__ void sort_kernel(const float* __restrict__ scores,
                            unsigned long long* __restrict__ sortbuf,
                            float* __restrict__ tk_score,
                            int*   __restrict__ tk_idx) {
  const int b = blockIdx.x;
  unsigned long long* buf = sortbuf + (size_t)b * SORTN;
  const int tid = threadIdx.x;  // 1024 threads
  for (int i = tid; i < SORTN; i += 1024) {
    unsigned long long v = 0ull;
    if (i < NTOT) {
      unsigned sb = __float_as_uint(scores[b * NTOT + i]);  // score > 0
      v = ((unsigned long long)sb << 32) | (unsigned)(~i);  // ~i: low index wins ties
    }
    buf[i] = v;
  }
  __syncthreads();
  for (int k = 2; k <= SORTN; k <<= 1) {
    for (int j = k >> 1; j > 0; j >>= 1) {
      for (int i = tid; i < SORTN; i += 1024) {
        int ixj = i ^ j;
        if (ixj > i) {
          unsigned long long a = buf[i], c = buf[ixj];
          bool sw = ((i & k) == 0) ? (a < c) : (a > c);  // overall descending
          if (sw) { buf[i] = c; buf[ixj] = a; }
        }
      }
      __syncthreads();
    }
  }
  for (int i = tid; i < KTOP; i += 1024) {
    unsigned long long v = buf[i];
    tk_score[b * KTOP + i] = __uint_as_float((unsigned)(v >> 32));
    tk_idx[b * KTOP + i]   = (int)(~(unsigned)v);
  }
}

// ---------------- Phase 3: gather + per-image max_coord + class offsets ---
__global__ void gather_kernel(const int* __restrict__ tk_idx,
                              const int* __restrict__ klass,
                              const float* __restrict__ boxes,
                              int*   __restrict__ tk_class,
                              float* __restrict__ tk_box,
                              float* __restrict__ off_box) {
  __shared__ float red[256];
  const int b = blockIdx.x, tid = threadIdx.x;
  float mx = -3.4e38f;
  for (int i = tid; i < KTOP; i += 256) {
    int idx = tk_idx[b * KTOP + i];
    const float* s4 = boxes + (size_t)(b * NTOT + idx) * 4;
    float x1 = s4[0], y1 = s4[1], x2 = s4[2], y2 = s4[3];
    tk_class[b * KTOP + i] = klass[b * NTOT + idx];
    float* d = tk_box + (size_t)(b * KTOP + i) * 4;
    d[0] = x1; d[1] = y1; d[2] = x2; d[3] = y2;
    mx = fmaxf(mx, fmaxf(fmaxf(x1, y1), fmaxf(x2, y2)));
  }
  red[tid] = mx;
  __syncthreads();
  for (int s = 128; s > 0; s >>= 1) {
    if (tid < s) red[tid] = fmaxf(red[tid], red[tid + s]);
    __syncthreads();
  }
  const float off_scale = red[0] + 1.0f;
  for (int i = tid; i < KTOP; i += 256) {
    float ofs = (float)tk_class[b * KTOP + i] * off_scale;
    const float* s4 = tk_box + (size_t)(b * KTOP + i) * 4;
    float* d = off_box + (size_t)(b * KTOP + i) * 4;
    d[0] = s4[0] + ofs; d[1] = s4[1] + ofs; d[2] = s4[2] + ofs; d[3] = s4[3] + ofs;
  }
}

// ---------------- Phase 4: pairwise IoU mask (j > i, iou > thr) -----------
__global__ void nms_mask_kernel(const float* __restrict__ off_box,
                                unsigned* __restrict__ mask) {
  __shared__ float sx1[KTOP], sy1[KTOP], sx2[KTOP], sy2[KTOP], sar[KTOP];
  const int b = blockIdx.y;
  const int tid = threadIdx.y * 32 + threadIdx.x;
  for (int i = tid; i < KTOP; i += 1024) {
    const float* s4 = off_box + (size_t)(b * KTOP + i) * 4;
    float x1 = s4[0], y1 = s4[1], x2 = s4[2], y2 = s4[3];
    sx1[i] = x1; sy1[i] = y1; sx2[i] = x2; sy2[i] = y2;
    sar[i] = (x2 - x1 + 1.0f) * (y2 - y1 + 1.0f);
  }
  __syncthreads();
  const int i = blockIdx.x * 32 + threadIdx.y;
  if (i >= KTOP) return;
  const float x1 = sx1[i], y1 = sy1[i], x2 = sx2[i], y2 = sy2[i], ai = sar[i];
  const int w = threadIdx.x;
  unsigned bits = 0;
  for (int jj = 0; jj < 32; ++jj) {
    int j = w * 32 + jj;
    if (j < KTOP && j > i) {
      float iw = fminf(x2, sx2[j]) - fmaxf(x1, sx1[j]);
      float ih = fminf(y2, sy2[j]) - fmaxf(y1, sy1[j]);
      float inter = fmaxf(iw, 0.0f) * fmaxf(ih, 0.0f);
      float iou = inter / (ai + sar[j] - inter);
      if (iou > IOU_THR) bits |= (1u << jj);
    }
  }
  mask[(size_t)(b * KTOP + i) * 32 + w] = bits;
}

// ---------------- Phase 5: serial suppression scan (one wave32 / image) ---
__global__ void nms_reduce_kernel(const unsigned* __restrict__ mask,
                                  const float* __restrict__ tk_score,
                                  int* __restrict__ keep) {
  const int b = blockIdx.x;
  const int lane = threadIdx.x;  // 32 lanes, lane owns mask word `lane`
  unsigned remv = 0;
  for (int i = 0; i < KTOP; ++i) {
    unsigned rw = __shfl(remv, i >> 5, 32);
    bool kept = (tk_score[b * KTOP + i] >= SCORE_THR) && (((rw >> (i & 31)) & 1u) == 0u);
    if (kept) remv |= mask[(size_t)(b * KTOP + i) * 32 + lane];
    if (lane == 0) keep[b * KTOP + i] = kept ? 1 : 0;
  }
}

// ---------------- Phase 6: emit dets | classes | keep ---------------------
__global__ void out_kernel(const float* __restrict__ tk_score,
                           const float* __restrict__ tk_box,
                           const int* __restrict__ tk_class,
                           const int* __restrict__ keep,
                           float* __restrict__ out) {
  const int g = blockIdx.x * blockDim.x + threadIdx.x;
  if (g >= NB * KTOP) return;
  const int k = keep[g];
  const float m = k ? 1.0f : 0.0f;
  const float* b4 = tk_box + (size_t)g * 4;
  float* d = out + (size_t)g * 5;
  d[0] = m * tk_score[g];
  d[1] = m * b4[0]; d[2] = m * b4[1]; d[3] = m * b4[2]; d[4] = m * b4[3];
  out[NB * KTOP * 5 + g] = (float)tk_class[g];
  out[NB * KTOP * 6 + g] = m;
}

// ---------------- host-side orchestration ---------------------------------
extern "C" void kernel_launch(void* const* d_in, const int* in_sizes, int n_in,
                              void* d_out, int out_size, void* d_ws, size_t ws_size,
                              hipStream_t stream) {
  (void)in_sizes; (void)n_in; (void)out_size; (void)ws_size;
  char* ws = (char*)d_ws;
  size_t off = 0;
  auto carve = [&](size_t bytes) -> void* {
    void* p = ws + off;
    off = (off + bytes + 255) & ~(size_t)255;
    return p;
  };
  float*              scores  = (float*)carve((size_t)NB * NTOT * 4);
  int*                klass   = (int*)  carve((size_t)NB * NTOT * 4);
  float*              boxes   = (float*)carve((size_t)NB * NTOT * 16);
  unsigned long long* sortbuf = (unsigned long long*)carve((size_t)NB * SORTN * 8);
  float*              tk_score= (float*)carve((size_t)NB * KTOP * 4);
  int*                tk_idx  = (int*)  carve((size_t)NB * KTOP * 4);
  int*                tk_class= (int*)  carve((size_t)NB * KTOP * 4);
  float*              tk_box  = (float*)carve((size_t)NB * KTOP * 16);
  float*              off_box = (float*)carve((size_t)NB * KTOP * 16);
  unsigned*           mask    = (unsigned*)carve((size_t)NB * KTOP * 32 * 4);
  int*                keep    = (int*)  carve((size_t)NB * KTOP * 4);

  struct Lvl { int H, W, stride, aoff; };
  const Lvl L[5] = {{100,128,8,0},{50,64,16,12800},{25,32,32,16000},
                    {13,16,64,16800},{7,8,128,17008}};
  for (int li = 0; li < 5; ++li) {
    const float* cls = (const float*)d_in[3 * li + 0];
    const float* cen = (const float*)d_in[3 * li + 1];
    const float* reg = (const float*)d_in[3 * li + 2];
    const int HW = L[li].H * L[li].W;
    const int tiles = (HW + 255) / 256;
    decode_kernel<<<dim3(NB * tiles), dim3(256), 0, stream>>>(
        cls, cen, reg, scores, klass, boxes, HW, L[li].W, L[li].stride,
        L[li].aoff, tiles);
  }
  sort_kernel<<<NB, 1024, 0, stream>>>(scores, sortbuf, tk_score, tk_idx);
  gather_kernel<<<NB, 256, 0, stream>>>(tk_idx, klass, boxes, tk_class, tk_box, off_box);
  nms_mask_kernel<<<dim3((KTOP + 31) / 32, NB), dim3(32, 32), 0, stream>>>(off_box, mask);
  nms_reduce_kernel<<<NB, 32, 0, stream>>>(mask, tk_score, keep);
  out_kernel<<<(NB * KTOP + 255) / 256, 256, 0, stream>>>(tk_score, tk_box, tk_class,
                                                          keep, (float*)d_out);
}